// Model_19756849561999
// MI455X (gfx1250) — compile-verified
//
#include <hip/hip_runtime.h>
#include <hip/hip_bf16.h>

#define FDIM 64

typedef __attribute__((ext_vector_type(2))) float v2f;
typedef __attribute__((ext_vector_type(8))) float v8f;

static inline int cdiv(long long a, long long b) { return (int)((a + b - 1) / b); }

// ---------------------------------------------------------------------------
// Degree / normalization
// ---------------------------------------------------------------------------
__global__ void deg_kernel(const long long* __restrict__ dst, float* __restrict__ deg, int nE) {
    int e = blockIdx.x * blockDim.x + threadIdx.x;
    if (e < nE) atomicAdd(&deg[(int)dst[e]], 1.0f);
}

__global__ void dinv_kernel(float* __restrict__ deg, int n) {
    int i = blockIdx.x * blockDim.x + threadIdx.x;
    if (i < n) deg[i] = rsqrtf(deg[i] + 1.0f);   // +1 = self loop; always > 0
}

// ---------------------------------------------------------------------------
// WMMA GEMM: out[M x 64] = A[M x 64] @ W[64 x 64] (+bias) (+relu)
// 256 threads = 8 wave32; wave w -> 16x16 tile (rowTile = w>>2, colTile = w&3)
// Block covers 32 rows x 64 cols. Uses V_WMMA_F32_16X16X4_F32 (fp32 exact-path).
// ---------------------------------------------------------------------------
#define SA_STRIDE 66   // padded LDS row stride (floats) to avoid bank conflicts

__global__ __launch_bounds__(256)
void gemm_n64_wmma(const float* __restrict__ A, const float* __restrict__ W,
                   const float* __restrict__ bias, float* __restrict__ out,
                   int M, int fuse)  // fuse bit0 = add bias, bit1 = relu
{
    __shared__ float sW[FDIM * FDIM];        // 16 KB
    __shared__ float sA[32 * SA_STRIDE];     // ~8.25 KB

    const int tid = threadIdx.x;
    const int rowBlock = blockIdx.x * 32;

    // Stage W (4096 floats) as float4: 4 per thread
    {
        const float4* Wv = (const float4*)W;
        float4* sWv = (float4*)sW;
#pragma unroll
        for (int i = 0; i < 4; ++i)
            sWv[tid + i * 256] = Wv[tid + i * 256];
    }
    // Stage A tile (32 x 64) scalar into padded LDS: 8 per thread
#pragma unroll
    for (int i = 0; i < 8; ++i) {
        int idx = tid + i * 256;              // 0..2047
        int r = idx >> 6, c = idx & 63;
        int gr = rowBlock + r;
        sA[r * SA_STRIDE + c] = (gr < M) ? A[(size_t)gr * FDIM + c] : 0.0f;
    }
    __syncthreads();

    const int wave = tid >> 5;
    const int lane = tid & 31;
    const int rowTile = (wave >> 2) * 16;     // 0 or 16
    const int colTile = (wave & 3) * 16;      // 0,16,32,48
    const int lhalf = lane & 15;
    const bool hi = (lane >= 16);

    const int arow = rowTile + lhalf;         // A row this lane supplies
    const int bcol = colTile + lhalf;         // B col this lane supplies

    v8f acc = {};
#pragma unroll
    for (int k = 0; k < 16; ++k) {
        const int kb = 4 * k + (hi ? 2 : 0);  // lanes 0-15: K=4k,4k+1; 16-31: 4k+2,4k+3
        v2f a = *(const v2f*)&sA[arow * SA_STRIDE + kb];   // 8B-aligned (kb even)
        v2f b;
        b.x = sW[(kb + 0) * FDIM + bcol];
        b.y = sW[(kb + 1) * FDIM + bcol];
        acc = __builtin_amdgcn_wmma_f32_16x16x4_f32(
            /*neg_a=*/false, a, /*neg_b=*/false, b,
            /*c_mod=*/(short)0, acc, /*reuse_a=*/false, /*reuse_b=*/false);
    }

    // D layout: VGPR i holds M = rowTile+i (lanes 0-15) / rowTile+8+i (lanes 16-31), N = lhalf
    const int colg = colTile + lhalf;
    float bv = (fuse & 1) ? bias[colg] : 0.0f;
#pragma unroll
    for (int i = 0; i < 8; ++i) {
        int r = rowBlock + rowTile + i + (hi ? 8 : 0);
        if (r < M) {
            float v = acc[i] + bv;
            if (fuse & 2) v = fmaxf(v, 0.0f);
            out[(size_t)r * FDIM + colg] = v;
        }
    }
}

// ---------------------------------------------------------------------------
// Edge scatter-add: acc[d,c] += t[s,c] * dinv[s]*dinv[d]
// One thread per (edge, channel): gathers are 256B coalesced per edge.
// ---------------------------------------------------------------------------
__global__ void agg_edge_kernel(const long long* __restrict__ se, const long long* __restrict__ de,
                                const float* __restrict__ dinv, const float* __restrict__ t,
                                float* __restrict__ acc, int nE)
{
    int tid = blockIdx.x * blockDim.x + threadIdx.x;
    int e = tid >> 6, c = tid & 63;
    if (e >= nE) return;
    int s = (int)se[e], d = (int)de[e];
    float nm = dinv[s] * dinv[d];
    atomicAdd(&acc[(size_t)d * FDIM + c], t[(size_t)s * FDIM + c] * nm);
}

// Self-loop + bias (+relu), in place on acc
__global__ void finalize_kernel(const float* __restrict__ t, const float* __restrict__ dinv,
                                const float* __restrict__ bias, float* __restrict__ acc,
                                int n, int relu)
{
    int tid = blockIdx.x * blockDim.x + threadIdx.x;
    int i = tid >> 6, c = tid & 63;
    if (i >= n) return;
    float di = dinv[i];
    float v = acc[tid] + t[tid] * di * di + bias[c];
    if (relu) v = fmaxf(v, 0.0f);
    acc[tid] = v;
}

// ---------------------------------------------------------------------------
// Global mean pool
// ---------------------------------------------------------------------------
__global__ void pool_kernel(const float* __restrict__ h, const long long* __restrict__ batch,
                            float* __restrict__ sums, float* __restrict__ cnt, int n)
{
    int tid = blockIdx.x * blockDim.x + threadIdx.x;
    int i = tid >> 6, c = tid & 63;
    if (i >= n) return;
    int g = (int)batch[i];
    atomicAdd(&sums[(size_t)g * FDIM + c], h[tid]);
    if (c == 0) atomicAdd(&cnt[g], 1.0f);
}

__global__ void mean_kernel(const float* __restrict__ sums, const float* __restrict__ cnt,
                            float* __restrict__ g, int ng)
{
    int tid = blockIdx.x * blockDim.x + threadIdx.x;
    int gi = tid >> 6;
    if (gi >= ng) return;
    g[tid] = sums[tid] / fmaxf(cnt[gi], 1.0f);
}

// pred[gi] = sum_c g3[gi,c]*W4[c] + b4
__global__ void pred_kernel(const float* __restrict__ g3, const float* __restrict__ W4,
                            const float* __restrict__ b4, float* __restrict__ out, int ng)
{
    int gi = blockIdx.x * blockDim.x + threadIdx.x;
    if (gi >= ng) return;
    float s = b4[0];
#pragma unroll 8
    for (int c = 0; c < FDIM; ++c) s += g3[(size_t)gi * FDIM + c] * W4[c];
    out[gi] = s;
}

// ---------------------------------------------------------------------------
// Launcher
// ---------------------------------------------------------------------------
extern "C" void kernel_launch(void* const* d_in, const int* in_sizes, int n_in,
                              void* d_out, int out_size, void* d_ws, size_t ws_size,
                              hipStream_t stream)
{
    const float*     x     = (const float*)d_in[0];
    const long long* eidx  = (const long long*)d_in[1];   // int64 [2, E]
    const long long* batch = (const long long*)d_in[2];   // int64 [N]
    const float* W1 = (const float*)d_in[3];
    const float* b1 = (const float*)d_in[4];
    const float* W2 = (const float*)d_in[5];
    const float* b2 = (const float*)d_in[6];
    const float* W3 = (const float*)d_in[7];
    const float* b3 = (const float*)d_in[8];
    const float* W4 = (const float*)d_in[9];
    const float* b4 = (const float*)d_in[10];
    float* out = (float*)d_out;

    const int n  = in_sizes[0] / FDIM;   // 100000
    const int nE = in_sizes[1] / 2;      // 1600000
    const int ng = out_size;             // 512

    const long long* src = eidx;
    const long long* dst = eidx + nE;

    // workspace layout (floats)
    float* ws   = (float*)d_ws;
    size_t nf   = (size_t)n * FDIM;
    float* bufT = ws;                       // GEMM outputs (t1, then t2)
    float* bufA = bufT + nf;                // aggregation accumulator / h
    float* dinv = bufA + nf;                // deg -> dinv in place
    float* sums = dinv + n;
    float* cnt  = sums + (size_t)ng * FDIM;
    float* gbuf = cnt + ng;
    float* g3   = gbuf + (size_t)ng * FDIM;

    const int TPB = 256;
    const int gN64 = cdiv((long long)n * FDIM, TPB);
    const int gE64 = cdiv((long long)nE * FDIM, TPB);

    // --- degree & normalization ---
    hipMemsetAsync(dinv, 0, (size_t)n * sizeof(float), stream);
    deg_kernel<<<cdiv(nE, TPB), TPB, 0, stream>>>(dst, dinv, nE);
    dinv_kernel<<<cdiv(n, TPB), TPB, 0, stream>>>(dinv, n);

    // --- layer 1: t1 = x @ W1 ; h1 = relu(agg(t1) + b1) ---
    gemm_n64_wmma<<<cdiv(n, 32), TPB, 0, stream>>>(x, W1, nullptr, bufT, n, 0);
    hipMemsetAsync(bufA, 0, nf * sizeof(float), stream);
    agg_edge_kernel<<<gE64, TPB, 0, stream>>>(src, dst, dinv, bufT, bufA, nE);
    finalize_kernel<<<gN64, TPB, 0, stream>>>(bufT, dinv, b1, bufA, n, 1);

    // --- layer 2: t2 = h1 @ W2 ; h2 = agg(t2) + b2 ---
    gemm_n64_wmma<<<cdiv(n, 32), TPB, 0, stream>>>(bufA, W2, nullptr, bufT, n, 0);
    hipMemsetAsync(bufA, 0, nf * sizeof(float), stream);
    agg_edge_kernel<<<gE64, TPB, 0, stream>>>(src, dst, dinv, bufT, bufA, nE);
    finalize_kernel<<<gN64, TPB, 0, stream>>>(bufT, dinv, b2, bufA, n, 0);

    // --- global mean pool ---
    hipMemsetAsync(sums, 0, (size_t)ng * FDIM * sizeof(float), stream);
    hipMemsetAsync(cnt,  0, (size_t)ng * sizeof(float), stream);
    pool_kernel<<<gN64, TPB, 0, stream>>>(bufA, batch, sums, cnt, n);
    mean_kernel<<<cdiv((long long)ng * FDIM, TPB), TPB, 0, stream>>>(sums, cnt, gbuf, ng);

    // --- head: g3 = relu(g @ W3 + b3) ; pred = g3 @ W4 + b4 ---
    gemm_n64_wmma<<<cdiv(ng, 32), TPB, 0, stream>>>(gbuf, W3, b3, g3, ng, 3);
    pred_kernel<<<cdiv(ng, TPB), TPB, 0, stream>>>(g3, W4, b4, out, ng);
}